// Model_5119601017092
// MI455X (gfx1250) — compile-verified
//
#include <hip/hip_runtime.h>
#include <hip/hip_bf16.h>

typedef __attribute__((ext_vector_type(2))) float v2f;
typedef __attribute__((ext_vector_type(8))) float v8f;

#define N_SRC1 500000
#define N_DST1 100000
#define N_DST2 20000
#define IN_F   128
#define H_F    128
#define N_CLS  16

// ---------------------------------------------------------------------------
// Zero workspace (accumulators + degree counters + h buffer)
// ---------------------------------------------------------------------------
__global__ void zero_ws_kernel(float4* __restrict__ p, size_t n4) {
    size_t i = (size_t)blockIdx.x * blockDim.x + threadIdx.x;
    size_t stride = (size_t)gridDim.x * blockDim.x;
    for (; i < n4; i += stride) p[i] = make_float4(0.f, 0.f, 0.f, 0.f);
}

// ---------------------------------------------------------------------------
// Edge scatter: one wave (32 lanes) per edge, 4 floats per lane (128 feats).
// acc[dst] += feat[src] * w ; deg[dst] += 1.  Accumulators are L2-resident.
// ---------------------------------------------------------------------------
__global__ void edge_scatter_kernel(const float* __restrict__ feat,
                                    const int*   __restrict__ src,
                                    const int*   __restrict__ dst,
                                    const float* __restrict__ w,
                                    float* __restrict__ acc,
                                    float* __restrict__ deg,
                                    int nEdges) {
    const int lane = threadIdx.x & 31;
    const int e = blockIdx.x * 8 + (threadIdx.x >> 5);
    if (e >= nEdges) return;                 // whole-wave uniform
    const int   s  = src[e];
    const int   d  = dst[e];
    const float we = w[e];
    const float4 xv = ((const float4*)(feat + (size_t)s * 128))[lane];
    float* ap = acc + (size_t)d * 128 + lane * 4;
    atomicAdd(ap + 0, xv.x * we);
    atomicAdd(ap + 1, xv.y * we);
    atomicAdd(ap + 2, xv.z * we);
    atomicAdd(ap + 3, xv.w * we);
    if (lane == 0) atomicAdd(deg + d, 1.0f);
}

// ---------------------------------------------------------------------------
// Layer 1 GEMM: h = relu([x_dst | acc/deg] @ [W_self; W_neigh] + b)
// M = N_DST1 (mult of 16), K = 256 (two 128 halves), N = 128.
// Block = 256 threads = 8 waves; block owns one 16-row M tile, each wave one
// 16-col N tile. v_wmma_f32_16x16x4_f32, K-loop of 64 WMMAs per wave.
// ---------------------------------------------------------------------------
__global__ void sage_gemm1_kernel(const float* __restrict__ x,
                                  const float* __restrict__ acc,
                                  const float* __restrict__ deg,
                                  const float* __restrict__ Ws,
                                  const float* __restrict__ Wn,
                                  const float* __restrict__ bias,
                                  float* __restrict__ h) {
    const int lane  = threadIdx.x & 31;
    const int wave  = threadIdx.x >> 5;
    const int m0    = blockIdx.x * 16;
    const int n0    = wave * 16;
    const int row   = m0 + (lane & 15);      // A-fragment row for this lane
    const int col   = n0 + (lane & 15);      // B/C column for this lane
    const int khalf = (lane >> 4) * 2;       // lanes 16-31 carry K+2,K+3

    const float* xr = x   + (size_t)row * IN_F;
    const float* ar = acc + (size_t)row * IN_F;
    const float invd = 1.0f / fmaxf(deg[row], 1.0f);

    v8f c = {};
    // self half: A = x rows, B = W_self (128x128)
    #pragma unroll 4
    for (int k = 0; k < IN_F; k += 4) {
        const int kk = k + khalf;
        v2f a, b;
        a.x = xr[kk];
        a.y = xr[kk + 1];
        b.x = Ws[(size_t)kk * H_F + col];
        b.y = Ws[(size_t)(kk + 1) * H_F + col];
        c = __builtin_amdgcn_wmma_f32_16x16x4_f32(false, a, false, b,
                                                  (short)0, c, false, false);
    }
    // neighbor half: A = (acc * invd) rows, B = W_neigh (128x128)
    #pragma unroll 4
    for (int k = 0; k < IN_F; k += 4) {
        const int kk = k + khalf;
        v2f a, b;
        a.x = ar[kk]     * invd;
        a.y = ar[kk + 1] * invd;
        b.x = Wn[(size_t)kk * H_F + col];
        b.y = Wn[(size_t)(kk + 1) * H_F + col];
        c = __builtin_amdgcn_wmma_f32_16x16x4_f32(false, a, false, b,
                                                  (short)0, c, false, false);
    }

    const float bv    = bias[col];
    const int   rbase = m0 + ((lane >> 4) * 8); // C: vgpr r -> M = r (+8 hi half)
    #pragma unroll
    for (int r = 0; r < 8; ++r) {
        float v = fmaxf(c[r] + bv, 0.0f);       // ReLU
        h[(size_t)(rbase + r) * H_F + col] = v;
    }
}

// ---------------------------------------------------------------------------
// Layer 2 GEMM + bias + row-wise log_softmax.
// M = N_DST2 (mult of 16), K = 256, N = 16 (single tile). One wave per M tile.
// Each output row lives in one C VGPR across a 16-lane half -> shfl_xor(…,16)
// reductions for max / sum-exp.
// ---------------------------------------------------------------------------
__global__ void sage_gemm2_kernel(const float* __restrict__ h,
                                  const float* __restrict__ acc,
                                  const float* __restrict__ deg,
                                  const float* __restrict__ Ws,
                                  const float* __restrict__ Wn,
                                  const float* __restrict__ bias,
                                  float* __restrict__ out) {
    const int lane = threadIdx.x & 31;
    const int wave = threadIdx.x >> 5;
    const int tile = blockIdx.x * 8 + wave;
    if (tile >= N_DST2 / 16) return;         // whole-wave uniform
    const int m0    = tile * 16;
    const int row   = m0 + (lane & 15);
    const int col   = lane & 15;
    const int khalf = (lane >> 4) * 2;

    const float* hr = h   + (size_t)row * H_F;
    const float* ar = acc + (size_t)row * H_F;
    const float invd = 1.0f / fmaxf(deg[row], 1.0f);

    v8f c = {};
    #pragma unroll 4
    for (int k = 0; k < H_F; k += 4) {
        const int kk = k + khalf;
        v2f a, b;
        a.x = hr[kk];
        a.y = hr[kk + 1];
        b.x = Ws[(size_t)kk * N_CLS + col];
        b.y = Ws[(size_t)(kk + 1) * N_CLS + col];
        c = __builtin_amdgcn_wmma_f32_16x16x4_f32(false, a, false, b,
                                                  (short)0, c, false, false);
    }
    #pragma unroll 4
    for (int k = 0; k < H_F; k += 4) {
        const int kk = k + khalf;
        v2f a, b;
        a.x = ar[kk]     * invd;
        a.y = ar[kk + 1] * invd;
        b.x = Wn[(size_t)kk * N_CLS + col];
        b.y = Wn[(size_t)(kk + 1) * N_CLS + col];
        c = __builtin_amdgcn_wmma_f32_16x16x4_f32(false, a, false, b,
                                                  (short)0, c, false, false);
    }

    const float bv    = bias[col];
    const int   rbase = m0 + ((lane >> 4) * 8);
    #pragma unroll
    for (int r = 0; r < 8; ++r) {
        float v = c[r] + bv;
        // max over the 16-lane group (one logits row)
        float m = v;
        m = fmaxf(m, __shfl_xor(m, 1, 16));
        m = fmaxf(m, __shfl_xor(m, 2, 16));
        m = fmaxf(m, __shfl_xor(m, 4, 16));
        m = fmaxf(m, __shfl_xor(m, 8, 16));
        float e = __expf(v - m);
        float s = e;
        s += __shfl_xor(s, 1, 16);
        s += __shfl_xor(s, 2, 16);
        s += __shfl_xor(s, 4, 16);
        s += __shfl_xor(s, 8, 16);
        out[(size_t)(rbase + r) * N_CLS + col] = (v - m) - __logf(s);
    }
}

// ---------------------------------------------------------------------------
// Host launcher
// ---------------------------------------------------------------------------
extern "C" void kernel_launch(void* const* d_in, const int* in_sizes, int n_in,
                              void* d_out, int out_size, void* d_ws, size_t ws_size,
                              hipStream_t stream) {
    const float* x    = (const float*)d_in[0];
    const int*   src1 = (const int*)  d_in[1];
    const int*   dst1 = (const int*)  d_in[2];
    const float* w1   = (const float*)d_in[3];
    const int*   src2 = (const int*)  d_in[4];
    const int*   dst2 = (const int*)  d_in[5];
    const float* w2   = (const float*)d_in[6];
    const float* Ws1  = (const float*)d_in[7];
    const float* Wn1  = (const float*)d_in[8];
    const float* b1   = (const float*)d_in[9];
    const float* Ws2  = (const float*)d_in[10];
    const float* Wn2  = (const float*)d_in[11];
    const float* b2   = (const float*)d_in[12];
    float* out = (float*)d_out;

    // Workspace layout (floats):
    float* acc1 = (float*)d_ws;                         // N_DST1*128
    float* deg1 = acc1 + (size_t)N_DST1 * 128;          // N_DST1
    float* hbuf = deg1 + N_DST1;                        // N_DST1*128
    float* acc2 = hbuf + (size_t)N_DST1 * 128;          // N_DST2*128
    float* deg2 = acc2 + (size_t)N_DST2 * 128;          // N_DST2
    const size_t totalFloats = (size_t)N_DST1 * 128 + N_DST1 +
                               (size_t)N_DST1 * 128 +
                               (size_t)N_DST2 * 128 + N_DST2;   // 28,280,000

    const int E1 = in_sizes[1];
    const int E2 = in_sizes[4];

    // 1) zero accumulators / degrees (h overwritten later, zeroing is cheap)
    zero_ws_kernel<<<2048, 256, 0, stream>>>((float4*)d_ws, totalFloats / 4);

    // 2) layer-1 edge scatter (wave per edge)
    edge_scatter_kernel<<<(E1 + 7) / 8, 256, 0, stream>>>(x, src1, dst1, w1,
                                                          acc1, deg1, E1);

    // 3) layer-1 fused GEMM + bias + ReLU  (WMMA f32 16x16x4)
    sage_gemm1_kernel<<<N_DST1 / 16, 256, 0, stream>>>(x, acc1, deg1,
                                                       Ws1, Wn1, b1, hbuf);

    // 4) layer-2 edge scatter
    edge_scatter_kernel<<<(E2 + 7) / 8, 256, 0, stream>>>(hbuf, src2, dst2, w2,
                                                          acc2, deg2, E2);

    // 5) layer-2 fused GEMM + bias + log_softmax  (WMMA f32 16x16x4)
    sage_gemm2_kernel<<<(N_DST2 / 16 + 7) / 8, 256, 0, stream>>>(hbuf, acc2, deg2,
                                                                 Ws2, Wn2, b2, out);
}